// RNAGNN_88622355185985
// MI455X (gfx1250) — compile-verified
//
#include <hip/hip_runtime.h>
#include <hip/hip_bf16.h>

// ---------------------------------------------------------------------------
// GAT stack (5 layers) for CDNA5 / gfx1250 (wave32).
//   gemm:       V_WMMA_F32_16X16X4_F32, wave per 16x16 tile, W staged in LDS,
//               branchless b64 A loads (out-of-range rows masked at store)
//   edge phase: 3 passes, f32 atomics in L2 (h fits in 192MB L2)
//   agg buffer is pre-initialized with the bias (out = segsum + b) so no
//   separate bias pass is needed.
// ---------------------------------------------------------------------------

#define GN 50000
#define GE 1250000
#define GC 64          // hidden / output channels (DIM == F_OUT == 64)
#define GEPS 1e-5f
#define GSLOPE 0.2f
#define WPITCH 80      // LDS pitch for W tile: 2*80 % 64 == 32 -> no bank clash

typedef __attribute__((ext_vector_type(2))) float v2f;
typedef __attribute__((ext_vector_type(4))) float v4f;
typedef __attribute__((ext_vector_type(8))) float v8f;

// ---------------- monotone float<->u32 encoding for atomic max -------------
__device__ __forceinline__ unsigned enc_f32(float f) {
  unsigned u = __float_as_uint(f);
  return (u & 0x80000000u) ? ~u : (u | 0x80000000u);
}
__device__ __forceinline__ float dec_f32(unsigned e) {
  unsigned u = (e & 0x80000000u) ? (e & 0x7FFFFFFFu) : ~e;
  return __uint_as_float(u);
}

// ---------------- WMMA GEMM: H[N x 64] = X[N x K] * W[K x 64] --------------
// block = 256 threads = 8 waves; each wave computes one 16x16 tile.
// Block tile: 32 rows x 64 cols. W (K x 64) staged in LDS once per block.
template <int K>
__global__ void __launch_bounds__(256)
gat_gemm_wmma(const float* __restrict__ X, const float* __restrict__ W,
              float* __restrict__ H, int nrows) {
  __shared__ float wlds[K * WPITCH];

  // cooperative stage of W into LDS (padded pitch), b128 loads
  {
    const int tid = threadIdx.x;
#pragma unroll
    for (int i = 0; i < (K * GC) / (256 * 4); ++i) {
      const int idx = (tid + i * 256) * 4;          // flat float index in W
      const int k = idx >> 6;                       // /64
      const int c = idx & 63;
      const v4f w = *(const v4f*)(W + idx);
      *(v4f*)(wlds + k * WPITCH + c) = w;
    }
  }
  __syncthreads();

  const int wave  = threadIdx.x >> 5;
  const int lane  = threadIdx.x & 31;
  const int lhalf = lane >> 4;       // 0: lanes 0-15, 1: lanes 16-31
  const int lmod  = lane & 15;
  const int rowT  = blockIdx.x * 32 + (wave >> 2) * 16;
  const int colT  = (wave & 3) * 16;

  const int row   = rowT + lmod;             // A-matrix M index for this lane
  const int rsafe = row < nrows ? row : 0;   // clamp; garbage masked at store
  const int col   = colT + lmod;             // B-matrix N index for this lane
  const float* __restrict__ xrow = X + (size_t)rsafe * K + lhalf * 2;

  v8f acc = {};
#pragma unroll
  for (int k0 = 0; k0 < K; k0 += 4) {
    const v2f a = *(const v2f*)(xrow + k0);            // one global_load_b64
    const int ka = k0 + lhalf * 2;
    v2f b;
    b.x = wlds[ka * WPITCH + col];
    b.y = wlds[(ka + 1) * WPITCH + col];
    acc = __builtin_amdgcn_wmma_f32_16x16x4_f32(
        /*neg_a=*/false, a, /*neg_b=*/false, b,
        /*c_mod=*/(short)0, acc, /*reuse_a=*/false, /*reuse_b=*/false);
  }

  // C/D layout: lanes 0-15: VGPR j = M=j ; lanes 16-31: VGPR j = M=8+j
  const int mbase = lhalf * 8;
#pragma unroll
  for (int j = 0; j < 8; ++j) {
    const int r = rowT + mbase + j;
    if (r < nrows) H[(size_t)r * GC + col] = acc[j];
  }
}

// ---------------- per-node attention scores: s = h.a_src, d = h.a_dst ------
__global__ void __launch_bounds__(256)
gat_scores(const float* __restrict__ H, const float* __restrict__ a_src,
           const float* __restrict__ a_dst, float* __restrict__ S,
           float* __restrict__ D, int n) {
  const int wave = threadIdx.x >> 5;
  const int lane = threadIdx.x & 31;
  const int node = blockIdx.x * 8 + wave;
  if (node >= n) return;
  const float* hr = H + (size_t)node * GC;
  float h0 = hr[lane], h1 = hr[lane + 32];
  float s = h0 * a_src[lane] + h1 * a_src[lane + 32];
  float d = h0 * a_dst[lane] + h1 * a_dst[lane + 32];
#pragma unroll
  for (int off = 16; off > 0; off >>= 1) {
    s += __shfl_xor(s, off, 32);
    d += __shfl_xor(d, off, 32);
  }
  if (lane == 0) { S[node] = s; D[node] = d; }
}

// ---------------- zero fill ------------------------------------------------
__global__ void gat_zero(float* __restrict__ p, long n) {
  long i = (long)blockIdx.x * blockDim.x + threadIdx.x;
  const long stride = (long)gridDim.x * blockDim.x;
  for (; i < n; i += stride) p[i] = 0.0f;
}

// ---------------- init agg buffer with bias (out = segsum + b) -------------
__global__ void gat_bias_init(float* __restrict__ X,
                              const float* __restrict__ b, long n) {
  long i = (long)blockIdx.x * blockDim.x + threadIdx.x;
  const long stride = (long)gridDim.x * blockDim.x;
  for (; i < n; i += stride) X[i] = b[i & (GC - 1)];
}

// ---------------- edge pass 1: leaky_relu score + segment max --------------
__global__ void __launch_bounds__(256)
gat_edge_score(const int* __restrict__ esrc, const int* __restrict__ edst,
               int ne, int n, const float* __restrict__ S,
               const float* __restrict__ D, float* __restrict__ escore,
               unsigned* __restrict__ m_enc) {
  const int e = blockIdx.x * blockDim.x + threadIdx.x;
  if (e >= ne + n) return;
  int si, di;
  if (e < ne) { si = esrc[e]; di = edst[e]; }
  else        { si = di = e - ne; }          // self loop
  float v = S[si] + D[di];
  v = (v > 0.0f) ? v : GSLOPE * v;           // leaky relu
  escore[e] = v;
  atomicMax(m_enc + di, enc_f32(v));
}

// ---------------- edge pass 2: exp(e - m) + segment sum --------------------
__global__ void __launch_bounds__(256)
gat_edge_exp(const int* __restrict__ edst, int ne, int n,
             const unsigned* __restrict__ m_enc, float* __restrict__ escore,
             float* __restrict__ z) {
  const int e = blockIdx.x * blockDim.x + threadIdx.x;
  if (e >= ne + n) return;
  const int di = (e < ne) ? edst[e] : e - ne;
  const float m  = dec_f32(m_enc[di]);
  const float ex = expf(escore[e] - m);
  escore[e] = ex;
  atomicAdd(z + di, ex);
}

// ---------------- edge pass 3: out[dst] += alpha * h[src] (wave per edge) --
__global__ void __launch_bounds__(256)
gat_edge_agg(const int* __restrict__ esrc, const int* __restrict__ edst,
             int ne, int n, const float* __restrict__ escore,
             const float* __restrict__ z, const float* __restrict__ H,
             float* __restrict__ out) {
  const int wave = threadIdx.x >> 5;
  const int lane = threadIdx.x & 31;
  const long e = (long)blockIdx.x * 8 + wave;
  if (e >= (long)ne + n) return;
  int si, di;
  if (e < ne) { si = esrc[e]; di = edst[e]; }
  else        { si = di = (int)(e - ne); }
  const float alpha = escore[e] / z[di];
  const float* hr = H + (size_t)si * GC;
  float* op = out + (size_t)di * GC;
  atomicAdd(op + lane,      alpha * hr[lane]);
  atomicAdd(op + lane + 32, alpha * hr[lane + 32]);
}

// ---------------- batch-norm statistics ------------------------------------
// stats[0..63] = sum_c, stats[64..127] = sumsq_c
__global__ void __launch_bounds__(256)
gat_bn_stats(const float* __restrict__ X, float* __restrict__ stats, int n) {
  __shared__ float ssum[256];
  __shared__ float ssq[256];
  const int c = threadIdx.x & (GC - 1);
  const int rlane = threadIdx.x >> 6;              // 4 rows in flight / block
  float s = 0.0f, q = 0.0f;
  for (int r = blockIdx.x * 4 + rlane; r < n; r += gridDim.x * 4) {
    const float v = X[(size_t)r * GC + c];
    s += v; q += v * v;
  }
  ssum[threadIdx.x] = s;
  ssq[threadIdx.x]  = q;
  __syncthreads();
  if (threadIdx.x < GC) {
    float ts = ssum[threadIdx.x] + ssum[threadIdx.x + 64] +
               ssum[threadIdx.x + 128] + ssum[threadIdx.x + 192];
    float tq = ssq[threadIdx.x] + ssq[threadIdx.x + 64] +
               ssq[threadIdx.x + 128] + ssq[threadIdx.x + 192];
    atomicAdd(stats + threadIdx.x, ts);
    atomicAdd(stats + GC + threadIdx.x, tq);
  }
}

// ---------------- batch-norm apply + relu (in place) -----------------------
__global__ void gat_bn_apply(float* __restrict__ X,
                             const float* __restrict__ stats,
                             const float* __restrict__ gamma,
                             const float* __restrict__ beta, int n) {
  long i = (long)blockIdx.x * blockDim.x + threadIdx.x;
  const long total = (long)n * GC;
  const long stride = (long)gridDim.x * blockDim.x;
  const float invn = 1.0f / (float)n;
  for (; i < total; i += stride) {
    const int c = (int)(i & (GC - 1));
    const float mu  = stats[c] * invn;
    const float var = stats[GC + c] * invn - mu * mu;
    float v = (X[i] - mu) * rsqrtf(var + GEPS) * gamma[c] + beta[c];
    X[i] = v > 0.0f ? v : 0.0f;
  }
}

// ---------------------------------------------------------------------------
// Host-side layer driver
// ---------------------------------------------------------------------------
static void run_gat_layer(const float* X, int K, const float* W,
                          const float* asrc, const float* adst,
                          const float* bias, const int* esrc, const int* edst,
                          float* h, float* S, float* D, unsigned* m_enc,
                          float* z, float* escore, float* agg,
                          hipStream_t stream) {
  const int n = GN, ne = GE, total = GE + GN;
  if (K == 128)
    gat_gemm_wmma<128><<<(n + 31) / 32, 256, 0, stream>>>(X, W, h, n);
  else
    gat_gemm_wmma<64><<<(n + 31) / 32, 256, 0, stream>>>(X, W, h, n);
  gat_scores<<<(n + 7) / 8, 256, 0, stream>>>(h, asrc, adst, S, D, n);
  gat_bias_init<<<2048, 256, 0, stream>>>(agg, bias, (long)n * GC);
  gat_zero<<<256, 256, 0, stream>>>(z, n);
  gat_zero<<<256, 256, 0, stream>>>((float*)m_enc, n);  // bits==0 sentinel
  gat_edge_score<<<(total + 255) / 256, 256, 0, stream>>>(esrc, edst, ne, n, S,
                                                          D, escore, m_enc);
  gat_edge_exp<<<(total + 255) / 256, 256, 0, stream>>>(edst, ne, n, m_enc,
                                                        escore, z);
  gat_edge_agg<<<(total + 7) / 8, 256, 0, stream>>>(esrc, edst, ne, n, escore,
                                                    z, h, agg);
}

extern "C" void kernel_launch(void* const* d_in, const int* in_sizes, int n_in,
                              void* d_out, int out_size, void* d_ws,
                              size_t ws_size, hipStream_t stream) {
  const float* x        = (const float*)d_in[0];   // [N,128]
  const int*   eidx     = (const int*)d_in[1];     // [2,E]
  const float* W_in     = (const float*)d_in[2];   // [128,64]
  const float* a_src_in = (const float*)d_in[3];
  const float* a_dst_in = (const float*)d_in[4];
  const float* b_in     = (const float*)d_in[5];
  const float* W_mid    = (const float*)d_in[6];   // [3,64,64]
  const float* a_src_m  = (const float*)d_in[7];   // [3,64]
  const float* a_dst_m  = (const float*)d_in[8];
  const float* b_mid    = (const float*)d_in[9];
  const float* gamma    = (const float*)d_in[10];  // [3,64]
  const float* beta     = (const float*)d_in[11];
  const float* W_out    = (const float*)d_in[12];  // [64,64]
  const float* a_src_o  = (const float*)d_in[13];
  const float* a_dst_o  = (const float*)d_in[14];
  const float* b_out    = (const float*)d_in[15];

  const int* esrc = eidx;
  const int* edst = eidx + GE;

  // workspace carve-up (floats)
  float* ws     = (float*)d_ws;
  float* h      = ws;                       // N*64
  float* bufA   = h + (long)GN * GC;        // N*64
  float* bufB   = bufA + (long)GN * GC;     // N*64
  float* S      = bufB + (long)GN * GC;     // N
  float* D      = S + GN;                   // N
  float* z      = D + GN;                   // N
  unsigned* m_e = (unsigned*)(z + GN);      // N
  float* escr   = (float*)(m_e + GN);       // E+N
  float* stats  = escr + (GE + GN);         // 128

  // ---- input layer: x[N,128] -> bufA ----
  run_gat_layer(x, 128, W_in, a_src_in, a_dst_in, b_in, esrc, edst, h, S, D,
                m_e, z, escr, bufA, stream);

  // ---- 3 middle layers with BN + ReLU ----
  float* cur = bufA;
  float* nxt = bufB;
  for (int l = 0; l < 3; ++l) {
    run_gat_layer(cur, GC, W_mid + (long)l * GC * GC, a_src_m + l * GC,
                  a_dst_m + l * GC, b_mid + l * GC, esrc, edst, h, S, D, m_e,
                  z, escr, nxt, stream);
    gat_zero<<<1, 128, 0, stream>>>(stats, 2 * GC);
    gat_bn_stats<<<1024, 256, 0, stream>>>(nxt, stats, GN);
    gat_bn_apply<<<2048, 256, 0, stream>>>(nxt, stats, gamma + l * GC,
                                           beta + l * GC, GN);
    float* t = cur; cur = nxt; nxt = t;
  }

  // ---- output layer -> d_out ----
  run_gat_layer(cur, GC, W_out, a_src_o, a_dst_o, b_out, esrc, edst, h, S, D,
                m_e, z, escr, (float*)d_out, stream);
}